// PlackettLuceLoss_1425929143041
// MI455X (gfx1250) — compile-verified
//
#include <hip/hip_runtime.h>
#include <hip/hip_bf16.h>
#include <math.h>

// Plackett-Luce NLL, B=1M rows x N=18. Memory-bound: ~226MB @ 23.3TB/s ~ 10us.
// Strategy: stage 256 rows/block into LDS via gfx1250 async global->LDS b128
// copies (perfectly coalesced 128-bit streams), one row per lane (wave32),
// sort-free O(N^2) suffix-logsumexp with unique stable keys, deterministic
// two-stage reduction (no float atomics).

#define NH     18
#define BLOCK  256
#define NEGF  (-1.0e30f)

#if defined(__has_builtin)
#  if __has_builtin(__builtin_amdgcn_global_load_async_to_lds_b128) && \
      __has_builtin(__builtin_amdgcn_s_wait_asynccnt)
#    define USE_ASYNC_LDS 1
#  endif
#endif
#ifndef USE_ASYNC_LDS
#  define USE_ASYNC_LDS 0
#endif

#if USE_ASYNC_LDS
typedef int v4i __attribute__((ext_vector_type(4)));
typedef __attribute__((address_space(1))) v4i as1_v4i;   // global
typedef __attribute__((address_space(3))) v4i as3_v4i;   // LDS
__device__ __forceinline__ void async_cp16(const void* g, void* l) {
    __builtin_amdgcn_global_load_async_to_lds_b128((as1_v4i*)g, (as3_v4i*)l, 0, 0);
}
#endif

__global__ __launch_bounds__(BLOCK) void pl_main_kernel(
    const float* __restrict__ scores,
    const int*   __restrict__ rankings,
    const int*   __restrict__ mask,
    float*       __restrict__ partial_loss,
    int*         __restrict__ partial_cnt,
    int B)
{
    __shared__ float sS[BLOCK * NH];
    __shared__ int   sR[BLOCK * NH];
    __shared__ int   sM[BLOCK * NH];

    const int  tid  = threadIdx.x;
    const int  row0 = blockIdx.x * BLOCK;
    const int  row  = row0 + tid;
    const bool full_block = (row0 + BLOCK) <= B;   // uniform per block

    float    sv[NH];
    unsigned key[NH];
    int      mv[NH];
    bool loaded = false;

#if USE_ASYNC_LDS
    if (full_block) {
        // Stage 256 rows of each array: 256*18*4 = 18432 B = 1152 x 16B chunks.
        const char* g0 = (const char*)(scores   + (size_t)row0 * NH);
        const char* g1 = (const char*)(rankings + (size_t)row0 * NH);
        const char* g2 = (const char*)(mask     + (size_t)row0 * NH);
        char* l0 = (char*)sS;
        char* l1 = (char*)sR;
        char* l2 = (char*)sM;
        const int bytes = BLOCK * NH * 4;
        for (int c = tid * 16; c < bytes; c += BLOCK * 16) {
            async_cp16(g0 + c, l0 + c);
            async_cp16(g1 + c, l1 + c);
            async_cp16(g2 + c, l2 + c);
        }
        __builtin_amdgcn_s_wait_asynccnt(0);   // this wave's async copies done
        __syncthreads();                        // all waves' copies visible
        // Row read: addr = tid*18 + i ; stride 18 (gcd(18,64)=2) -> the 32
        // lanes of a wave hit 32 distinct (even) banks: conflict-free.
        #pragma unroll
        for (int i = 0; i < NH; ++i) {
            float s = sS[tid * NH + i];
            int   r = sR[tid * NH + i];
            int   m = sM[tid * NH + i];
            mv[i]  = m;
            sv[i]  = m ? s : NEGF;
            key[i] = m ? (((unsigned)r << 5) | (unsigned)i)
                       : (0xFFFFFFE0u | (unsigned)i);
        }
        loaded = true;
    }
#endif
    if (!loaded) {
        if (row < B) {
            const float* sp = scores   + (size_t)row * NH;
            const int*   rp = rankings + (size_t)row * NH;
            const int*   mp = mask     + (size_t)row * NH;
            #pragma unroll
            for (int i = 0; i < NH; ++i) {
                int m  = mp[i];
                mv[i]  = m;
                sv[i]  = m ? sp[i] : NEGF;
                key[i] = m ? (((unsigned)rp[i] << 5) | (unsigned)i)
                           : (0xFFFFFFE0u | (unsigned)i);
            }
        } else {
            #pragma unroll
            for (int i = 0; i < NH; ++i) {
                mv[i] = 0; sv[i] = NEGF; key[i] = 0xFFFFFFE0u | (unsigned)i;
            }
        }
    }

    // Row-global max (== NEGF when k==0, matching the reference exactly).
    float M = sv[0];
    #pragma unroll
    for (int i = 1; i < NH; ++i) M = fmaxf(M, sv[i]);

    // k = #valid; maxvk = max key among valid (keys are unique).
    int k = 0; unsigned maxvk = 0u;
    #pragma unroll
    for (int i = 0; i < NH; ++i) {
        k += mv[i];
        unsigned cand = mv[i] ? key[i] : 0u;
        maxvk = (cand > maxvk) ? cand : maxvk;
    }

    // exp(s - M); masked lanes give exp(-1e30 - M) == 0 exactly (as in ref).
    float e[NH];
    #pragma unroll
    for (int i = 0; i < NH; ++i) e[i] = expf(sv[i] - M);

    // For each valid, non-last horse h: suffix set = { g : K_g >= K_h }.
    float loss = 0.0f;
    #pragma unroll
    for (int h = 0; h < NH; ++h) {
        float C = 0.0f;
        #pragma unroll
        for (int g = 0; g < NH; ++g) C += (key[g] >= key[h]) ? e[g] : 0.0f;
        bool take = (mv[h] != 0) && (key[h] != maxvk);
        loss += take ? (M + logf(C) - sv[h]) : 0.0f;
    }
    int race = (k > 1) ? 1 : 0;

    // wave32 reduction, then per-block partial (deterministic: no atomics).
    #pragma unroll
    for (int off = 16; off > 0; off >>= 1) {
        loss += __shfl_down(loss, off, 32);
        race += __shfl_down(race, off, 32);
    }
    __shared__ float wl[BLOCK / 32];
    __shared__ int   wr[BLOCK / 32];
    const int lane = tid & 31, wid = tid >> 5;
    if (lane == 0) { wl[wid] = loss; wr[wid] = race; }
    __syncthreads();
    if (tid == 0) {
        float L = 0.0f; int R = 0;
        #pragma unroll
        for (int w = 0; w < BLOCK / 32; ++w) { L += wl[w]; R += wr[w]; }
        partial_loss[blockIdx.x] = L;
        partial_cnt[blockIdx.x]  = R;
    }
}

__global__ __launch_bounds__(256) void pl_finalize_kernel(
    const float* __restrict__ partial_loss,
    const int*   __restrict__ partial_cnt,
    int nblocks, float* __restrict__ out)
{
    __shared__ float sl[256];
    __shared__ int   sc[256];
    float L = 0.0f; int C = 0;
    for (int i = threadIdx.x; i < nblocks; i += 256) {  // fixed order: deterministic
        L += partial_loss[i];
        C += partial_cnt[i];
    }
    sl[threadIdx.x] = L; sc[threadIdx.x] = C;
    __syncthreads();
    #pragma unroll
    for (int s = 128; s > 0; s >>= 1) {
        if (threadIdx.x < s) {
            sl[threadIdx.x] += sl[threadIdx.x + s];
            sc[threadIdx.x] += sc[threadIdx.x + s];
        }
        __syncthreads();
    }
    if (threadIdx.x == 0) {
        int denom = sc[0] > 1 ? sc[0] : 1;
        out[0] = sl[0] / (float)denom;
    }
}

extern "C" void kernel_launch(void* const* d_in, const int* in_sizes, int n_in,
                              void* d_out, int out_size, void* d_ws, size_t ws_size,
                              hipStream_t stream) {
    const float* scores   = (const float*)d_in[0];
    const int*   rankings = (const int*)d_in[1];
    const int*   mask     = (const int*)d_in[2];
    const int    B        = in_sizes[0] / NH;        // 1048576
    const int    nblocks  = (B + BLOCK - 1) / BLOCK; // 4096

    float* partial_loss = (float*)d_ws;
    int*   partial_cnt  = (int*)((char*)d_ws + (size_t)nblocks * sizeof(float));

    pl_main_kernel<<<nblocks, BLOCK, 0, stream>>>(scores, rankings, mask,
                                                  partial_loss, partial_cnt, B);
    pl_finalize_kernel<<<1, 256, 0, stream>>>(partial_loss, partial_cnt,
                                              nblocks, (float*)d_out);
}